// BalanceLoss_24129126269544
// MI455X (gfx1250) — compile-verified
//
#include <hip/hip_runtime.h>
#include <stdint.h>

// ---------------------------------------------------------------------------
// BalanceLoss (OHEM BCE) for MI455X / gfx1250.
//   N = 26.2M fp32 elements per tensor (pred, gt, mask). Output: 1 fp32 scalar.
// Algorithm: 2 streaming passes + histogram select (no sort).
//   Pass1: pos/neg counts (exact u32), f64 pos-loss sum, 2^18-bin histogram of
//          negative-loss float bits. Streaming uses the gfx1250 async
//          global->LDS engine with double-buffered tiles and s_wait_asynccnt
//          pipelining (ASYNCcnt completes in issue order).
//   Select: suffix scan -> threshold bin b*, count_above, k = min(neg, 3*pos).
//   Pass2: f64 sums of losses in bins > b* and == b*.
//   Final: topk = sum_gt + (k - count_above) * mean(bin b*); balanced scalar.
// ---------------------------------------------------------------------------

#define NBINS (1u << 18)          // bins over float bits >> 14 (rel width 2^-9)
#define BIN_SHIFT 14

struct Scalars {
  unsigned long long pos_cnt;
  unsigned long long neg_cnt;
  double pos_loss_sum;
  double sum_gt;          // sum of neg losses in bins strictly above b*
  double sum_eq;          // sum of neg losses in bin == b*
  unsigned long long k;
  unsigned long long count_above;
  unsigned long long count_eq;
  unsigned int b_star;
  unsigned int pad;
};

// ---- CDNA5 async global->LDS copy (ASYNCcnt path) --------------------------
__device__ __forceinline__ void async_copy_b128(unsigned lds_off, const void* gaddr) {
  unsigned long long a = (unsigned long long)(uintptr_t)gaddr;
  asm volatile("global_load_async_to_lds_b128 %0, %1, off"
               :: "v"(lds_off), "v"(a) : "memory");
}
// ASYNCcnt decrements in issue order for loads: waiting <=3 with 3 newly issued
// loads outstanding guarantees the previous tile's 3 loads have completed.
__device__ __forceinline__ void wait_asynccnt0() {
  asm volatile("s_wait_asynccnt 0" ::: "memory");
}
__device__ __forceinline__ void wait_asynccnt3() {
  asm volatile("s_wait_asynccnt 3" ::: "memory");
}

// ---- wave32 / block reductions --------------------------------------------
__device__ __forceinline__ double waveReduceD(double v) {
  for (int o = 16; o > 0; o >>= 1) v += __shfl_down(v, o, 32);
  return v;
}
__device__ __forceinline__ unsigned waveReduceU(unsigned v) {
  for (int o = 16; o > 0; o >>= 1) v += __shfl_down(v, o, 32);
  return v;
}
__device__ __forceinline__ double blockReduceD(double v, double* s) {
  const int lane = threadIdx.x & 31, wid = threadIdx.x >> 5;
  v = waveReduceD(v);
  if (lane == 0) s[wid] = v;
  __syncthreads();
  double r = 0.0;
  if (wid == 0) {
    const int nw = (blockDim.x + 31) >> 5;
    r = (lane < nw) ? s[lane] : 0.0;
    r = waveReduceD(r);
  }
  __syncthreads();
  return r;  // valid in thread 0
}
__device__ __forceinline__ unsigned blockReduceU(unsigned v, unsigned* s) {
  const int lane = threadIdx.x & 31, wid = threadIdx.x >> 5;
  v = waveReduceU(v);
  if (lane == 0) s[wid] = v;
  __syncthreads();
  unsigned r = 0;
  if (wid == 0) {
    const int nw = (blockDim.x + 31) >> 5;
    r = (lane < nw) ? s[lane] : 0u;
    r = waveReduceU(r);
  }
  __syncthreads();
  return r;  // valid in thread 0
}

// BCE loss of one element; gt, mask are exactly 0.0 or 1.0.
// loss = -(gt*max(log p,-100) + (1-gt)*max(log(1-p),-100)) = min(-log(t),100)
__device__ __forceinline__ float bce_loss(float p, bool isPos) {
  float t = isPos ? p : (1.0f - p);
  return fminf(-logf(t), 100.0f);
}

// ---------------------------------------------------------------------------
__global__ void __launch_bounds__(256)
pass1_kernel(const float* __restrict__ pred, const float* __restrict__ gt,
             const float* __restrict__ mask, long long N,
             unsigned* __restrict__ hist, Scalars* sc) {
  __shared__ float4 sP[2][256], sG[2][256], sM[2][256];  // double-buffered tiles
  __shared__ double dred[8];
  __shared__ unsigned ured[8];

  const int lane = threadIdx.x & 31;
  const long long wavesPerBlock = blockDim.x >> 5;
  const long long gwave = (long long)blockIdx.x * wavesPerBlock + (threadIdx.x >> 5);
  const long long nwaves = (long long)gridDim.x * wavesPerBlock;
  const long long ntiles = N >> 7;               // 128 floats per wave-tile

  unsigned pos_c = 0, neg_c = 0;
  double pos_l = 0.0;

  const unsigned ldsP[2] = {(unsigned)(uintptr_t)&sP[0][threadIdx.x],
                            (unsigned)(uintptr_t)&sP[1][threadIdx.x]};
  const unsigned ldsG[2] = {(unsigned)(uintptr_t)&sG[0][threadIdx.x],
                            (unsigned)(uintptr_t)&sG[1][threadIdx.x]};
  const unsigned ldsM[2] = {(unsigned)(uintptr_t)&sM[0][threadIdx.x],
                            (unsigned)(uintptr_t)&sM[1][threadIdx.x]};

  // ---- software-pipelined async streaming (wave-synchronous, EXEC all-1s) --
  long long t = gwave;
  int cur = 0;
  if (t < ntiles) {
    const long long v4 = (t << 5) + lane;
    async_copy_b128(ldsP[0], (const float4*)pred + v4);
    async_copy_b128(ldsG[0], (const float4*)gt + v4);
    async_copy_b128(ldsM[0], (const float4*)mask + v4);
  }
  for (; t < ntiles; t += nwaves, cur ^= 1) {
    const long long tn = t + nwaves;
    if (tn < ntiles) {                           // prefetch next tile, then
      const long long v4n = (tn << 5) + lane;    // wait only for current tile
      async_copy_b128(ldsP[cur ^ 1], (const float4*)pred + v4n);
      async_copy_b128(ldsG[cur ^ 1], (const float4*)gt + v4n);
      async_copy_b128(ldsM[cur ^ 1], (const float4*)mask + v4n);
      wait_asynccnt3();
    } else {
      wait_asynccnt0();
    }
    const float4 p4 = sP[cur][threadIdx.x];
    const float4 g4 = sG[cur][threadIdx.x];
    const float4 m4 = sM[cur][threadIdx.x];
    const float pe[4] = {p4.x, p4.y, p4.z, p4.w};
    const float ge[4] = {g4.x, g4.y, g4.z, g4.w};
    const float me[4] = {m4.x, m4.y, m4.z, m4.w};
#pragma unroll
    for (int j = 0; j < 4; ++j) {
      if (me[j] != 0.0f) {
        const bool isPos = (ge[j] != 0.0f);
        const float L = bce_loss(pe[j], isPos);
        if (isPos) { pos_c++; pos_l += (double)L; }
        else {
          neg_c++;
          atomicAdd(&hist[__float_as_uint(L) >> BIN_SHIFT], 1u);
        }
      }
    }
  }

  // tail (N not multiple of 128): scalar path
  const long long gtid = (long long)blockIdx.x * blockDim.x + threadIdx.x;
  const long long gstride = (long long)gridDim.x * blockDim.x;
  for (long long i = (ntiles << 7) + gtid; i < N; i += gstride) {
    const float m = mask[i];
    if (m != 0.0f) {
      const bool isPos = (gt[i] != 0.0f);
      const float L = bce_loss(pred[i], isPos);
      if (isPos) { pos_c++; pos_l += (double)L; }
      else { neg_c++; atomicAdd(&hist[__float_as_uint(L) >> BIN_SHIFT], 1u); }
    }
  }

  const double bl = blockReduceD(pos_l, dred);
  const unsigned bp = blockReduceU(pos_c, ured);
  const unsigned bn = blockReduceU(neg_c, ured);
  if (threadIdx.x == 0) {
    atomicAdd(&sc->pos_loss_sum, bl);
    atomicAdd(&sc->pos_cnt, (unsigned long long)bp);
    atomicAdd(&sc->neg_cnt, (unsigned long long)bn);
  }
}

// ---------------------------------------------------------------------------
__global__ void __launch_bounds__(256)
select_kernel(const unsigned* __restrict__ hist, Scalars* sc) {
  __shared__ unsigned long long chunk[256];
  __shared__ unsigned long long suffix[256];
  __shared__ unsigned long long k_sh;
  const int t = threadIdx.x;
  const int CHUNK = NBINS / 256;                 // 1024 bins per thread

  unsigned long long s = 0;
  for (int b = t * CHUNK; b < (t + 1) * CHUNK; ++b) s += hist[b];
  chunk[t] = s;
  __syncthreads();

  if (t == 0) {
    const unsigned long long pos = sc->pos_cnt;
    const unsigned long long neg = sc->neg_cnt;
    unsigned long long k = 3ull * pos;           // NEG_RATIO = 3
    if (neg < k) k = neg;
    sc->k = k; k_sh = k;
    unsigned long long run = 0;                  // exclusive suffix over chunks
    for (int c = 255; c >= 0; --c) { suffix[c] = run; run += chunk[c]; }
    sc->b_star = 0xFFFFFFFFu;                    // sentinel: nothing qualifies
    sc->count_above = 0ull;
    sc->count_eq = 1ull;                         // avoid div-by-0
  }
  __syncthreads();

  const unsigned long long k = k_sh;
  if (k > 0) {
    unsigned long long run = suffix[t];          // count in all bins above chunk
    for (int b = (t + 1) * CHUNK - 1; b >= t * CHUNK; --b) {
      const unsigned long long c = hist[b];
      if (run < k && run + c >= k) {             // exactly one thread hits this
        sc->b_star = (unsigned)b;
        sc->count_above = run;
        sc->count_eq = c;
      }
      run += c;
    }
  }
}

// ---------------------------------------------------------------------------
__global__ void __launch_bounds__(256)
pass2_kernel(const float* __restrict__ pred, const float* __restrict__ gt,
             const float* __restrict__ mask, long long N, Scalars* sc) {
  __shared__ double dred[8];
  if (sc->k == 0) return;                        // fallback path: nothing to do
  const unsigned bstar = sc->b_star;

  double s_gt = 0.0, s_eq = 0.0;
  const long long NV4 = N >> 2;
  const long long gtid = (long long)blockIdx.x * blockDim.x + threadIdx.x;
  const long long gstride = (long long)gridDim.x * blockDim.x;
  const float4* p4 = (const float4*)pred;
  const float4* g4 = (const float4*)gt;
  const float4* m4 = (const float4*)mask;

  for (long long i = gtid; i < NV4; i += gstride) {
    __builtin_prefetch(p4 + i + gstride, 0, 1);  // global_prefetch_b8
    const float4 p = p4[i], g = g4[i], m = m4[i];
    const float pe[4] = {p.x, p.y, p.z, p.w};
    const float ge[4] = {g.x, g.y, g.z, g.w};
    const float me[4] = {m.x, m.y, m.z, m.w};
#pragma unroll
    for (int j = 0; j < 4; ++j) {
      if (me[j] != 0.0f && ge[j] == 0.0f) {
        const float L = fminf(-logf(1.0f - pe[j]), 100.0f);
        const unsigned bin = __float_as_uint(L) >> BIN_SHIFT;
        if (bin > bstar)       s_gt += (double)L;
        else if (bin == bstar) s_eq += (double)L;
      }
    }
  }
  for (long long i = (NV4 << 2) + gtid; i < N; i += gstride) {
    if (mask[i] != 0.0f && gt[i] == 0.0f) {
      const float L = fminf(-logf(1.0f - pred[i]), 100.0f);
      const unsigned bin = __float_as_uint(L) >> BIN_SHIFT;
      if (bin > bstar)       s_gt += (double)L;
      else if (bin == bstar) s_eq += (double)L;
    }
  }

  const double bg = blockReduceD(s_gt, dred);
  const double be = blockReduceD(s_eq, dred);
  if (threadIdx.x == 0) {
    atomicAdd(&sc->sum_gt, bg);
    atomicAdd(&sc->sum_eq, be);
  }
}

// ---------------------------------------------------------------------------
__global__ void finalize_kernel(const Scalars* sc, float* out) {
  const double posl = sc->pos_loss_sum;
  const double posc = (double)sc->pos_cnt;
  const unsigned long long k = sc->k;
  double res;
  if (k > 0) {
    const double rem = (double)(k - sc->count_above);
    const double mean_eq = sc->sum_eq / (double)sc->count_eq;
    const double topk = sc->sum_gt + rem * mean_eq;
    res = (posl + topk) / (posc + (double)k + 1e-6);
  } else {
    res = posl / (posc + 1e-6);
  }
  out[0] = (float)res;
}

// ---------------------------------------------------------------------------
extern "C" void kernel_launch(void* const* d_in, const int* in_sizes, int n_in,
                              void* d_out, int out_size, void* d_ws, size_t ws_size,
                              hipStream_t stream) {
  (void)n_in; (void)out_size; (void)ws_size;
  const float* pred = (const float*)d_in[0];
  const float* gt   = (const float*)d_in[1];
  const float* mask = (const float*)d_in[2];
  const long long N = (long long)in_sizes[0];

  unsigned* hist = (unsigned*)d_ws;
  Scalars* sc = (Scalars*)((char*)d_ws + (size_t)NBINS * sizeof(unsigned));

  hipMemsetAsync(d_ws, 0, (size_t)NBINS * sizeof(unsigned) + sizeof(Scalars), stream);

  const int threads = 256;                       // 8 wave32s per block
  const int blocks = 2048;
  pass1_kernel<<<blocks, threads, 0, stream>>>(pred, gt, mask, N, hist, sc);
  select_kernel<<<1, 256, 0, stream>>>(hist, sc);
  pass2_kernel<<<blocks, threads, 0, stream>>>(pred, gt, mask, N, sc);
  finalize_kernel<<<1, 1, 0, stream>>>(sc, (float*)d_out);
}